// OptimizedTradingLSTM_377957122433
// MI455X (gfx1250) — compile-verified
//
#include <hip/hip_runtime.h>
#include <math.h>

// ---------------------------------------------------------------------------
// OptimizedTradingLSTM forward for MI455X (gfx1250, wave32, WMMA f32 16x16x4).
// All matmuls (projections, conv-taps-as-GEMM, LSTM input+recurrent, attention
// QKV/out, heads) run through one WMMA GEMM kernel. Inner loops are pure
// load->wmma (no masks, no branches); masks appear only where semantically
// required (conv sequence padding on A, ragged-K tail). Attention core is a
// fused flash-style LDS kernel. Param leaves assumed in jax pytree
// (sorted-key) flatten order; x detected by its unique size.
// ---------------------------------------------------------------------------

#define BATCH 256
#define SEQ 100
#define NFEAT 74
#define HDIM 256
#define NHEADS 8
#define HEADDIM 32
#define MSROWS (BATCH * SEQ) /* 25600 */
#define EPSN 1e-5f

typedef float v2f __attribute__((ext_vector_type(2)));
typedef float v8f __attribute__((ext_vector_type(8)));

__device__ __forceinline__ float gelu_f(float x) {
  return 0.5f * x * (1.0f + erff(x * 0.70710678118654752f));
}
__device__ __forceinline__ float sigm_f(float x) {
  return 1.0f / (1.0f + expf(-x));
}

// ---------------------------------------------------------------------------
// Generic GEMM: C[M,N] (+)= A[M,K] * B (+bias) with epilogue activation.
//  opB==0 : B is KxN row-major (ldb = row stride)
//  opB==1 : B is NxK row-major (computes A @ B^T, contiguous K loads)
//  seqLen>0: A row m is remapped to (m/seqLen, m%seqLen + shiftS) with zero
//            padding outside [0,seqLen) -- dilated conv taps.
// One wave computes one 16x16 tile via V_WMMA_F32_16X16X4_F32.
// Out-of-range lanes load from clamped in-bounds addresses; because WMMA rows
// and columns are independent, garbage only reaches unstored D rows/cols, so
// no masking is needed in the main loops (stores are guarded). The A-side
// mask is applied only for conv sequence padding; the K tail masks A only.
// ---------------------------------------------------------------------------
__global__ void k_gemm(const float* __restrict__ A, const float* __restrict__ Bm,
                       const float* __restrict__ bias, float* __restrict__ C,
                       int M, int N, int K, int lda, int ldb, int ldc,
                       int opB, int accumulate, int act, int shiftS, int seqLen) {
  const int wave = threadIdx.x >> 5;
  const int lane = threadIdx.x & 31;
  const int tile = blockIdx.x * (blockDim.x >> 5) + wave;
  const int tnCnt = (N + 15) >> 4;
  const int tm = tile / tnCnt;
  const int tn = tile % tnCnt;
  const int r = lane & 15;
  const int half = lane >> 4;

  const int m = tm * 16 + r;  // A row owned by this lane
  const int n = tn * 16 + r;  // B column owned by this lane

  // Resolve (possibly sequence-shifted) A row; clamp to a valid address.
  int srcRow = m;
  bool rowValid = (m < M);
  if (seqLen > 0) {
    int bb = m / seqLen;
    int ss = (m % seqLen) + shiftS;
    rowValid = rowValid && (ss >= 0) && (ss < seqLen);
    srcRow = bb * seqLen + ss;
  }
  const float maskA = rowValid ? 1.0f : 0.0f;
  const int rowC = rowValid ? srcRow : 0;
  const int nC = (n < N) ? n : 0;

  const float* __restrict__ Ap = A + (size_t)rowC * lda;
  const float* __restrict__ Bp = opB ? (Bm + (size_t)nC * ldb) : (Bm + nC);

  v8f acc = {};
  const int kMain = K & ~3;
  if (seqLen == 0) {
    // Pure load -> wmma; no masks needed (row/col independence + store guards).
    if (opB) {
#pragma unroll 4
      for (int kb = 0; kb < kMain; kb += 4) {
        const int k0 = kb + 2 * half;
        v2f a, b;
        a.x = Ap[k0];
        a.y = Ap[k0 + 1];
        b.x = Bp[k0];
        b.y = Bp[k0 + 1];
        acc = __builtin_amdgcn_wmma_f32_16x16x4_f32(false, a, false, b, (short)0,
                                                    acc, false, false);
      }
    } else {
#pragma unroll 4
      for (int kb = 0; kb < kMain; kb += 4) {
        const int k0 = kb + 2 * half;
        v2f a, b;
        a.x = Ap[k0];
        a.y = Ap[k0 + 1];
        b.x = Bp[(size_t)k0 * ldb];
        b.y = Bp[(size_t)(k0 + 1) * ldb];
        acc = __builtin_amdgcn_wmma_f32_16x16x4_f32(false, a, false, b, (short)0,
                                                    acc, false, false);
      }
    }
  } else {
    // Conv-tap path: zero contribution for out-of-sequence source rows.
    if (opB) {
#pragma unroll 4
      for (int kb = 0; kb < kMain; kb += 4) {
        const int k0 = kb + 2 * half;
        v2f a, b;
        a.x = maskA * Ap[k0];
        a.y = maskA * Ap[k0 + 1];
        b.x = Bp[k0];
        b.y = Bp[k0 + 1];
        acc = __builtin_amdgcn_wmma_f32_16x16x4_f32(false, a, false, b, (short)0,
                                                    acc, false, false);
      }
    } else {
#pragma unroll 4
      for (int kb = 0; kb < kMain; kb += 4) {
        const int k0 = kb + 2 * half;
        v2f a, b;
        a.x = maskA * Ap[k0];
        a.y = maskA * Ap[k0 + 1];
        b.x = Bp[(size_t)k0 * ldb];
        b.y = Bp[(size_t)(k0 + 1) * ldb];
        acc = __builtin_amdgcn_wmma_f32_16x16x4_f32(false, a, false, b, (short)0,
                                                    acc, false, false);
      }
    }
  }
  if (kMain < K) {  // masked tail for K % 4 != 0 (small projection GEMMs)
    const int k0 = kMain + 2 * half;
    const int k1 = k0 + 1;
    const float ka = ((k0 < K) ? 1.0f : 0.0f) * maskA;
    const float kb1 = ((k1 < K) ? 1.0f : 0.0f) * maskA;
    const int k0c = (k0 < K) ? k0 : 0;
    const int k1c = (k1 < K) ? k1 : 0;
    v2f a, b;
    a.x = ka * Ap[k0c];
    a.y = kb1 * Ap[k1c];
    if (opB) {
      b.x = Bp[k0c];
      b.y = Bp[k1c];
    } else {
      b.x = Bp[(size_t)k0c * ldb];
      b.y = Bp[(size_t)k1c * ldb];
    }
    acc = __builtin_amdgcn_wmma_f32_16x16x4_f32(false, a, false, b, (short)0,
                                                acc, false, false);
  }

  // Epilogue (cold; guards fine here).
  const int col = tn * 16 + r;
  const float bval = (bias != nullptr && col < N) ? bias[col] : 0.0f;
#pragma unroll
  for (int v = 0; v < 8; ++v) {
    const int row = tm * 16 + v + 8 * half;  // C layout: VGPR v, halves split M
    if (row < M && col < N) {
      float x = acc[v] + bval;
      float* cp = &C[(size_t)row * ldc + col];
      if (accumulate) x += *cp;
      if (act == 1) x = gelu_f(x);
      else if (act == 2) x = sigm_f(x);
      *cp = x;
    }
  }
}

// --------------------------- elementwise / reductions ----------------------
__global__ void k_featnorm(const float* __restrict__ x, const float* rm,
                           const float* rv, const float* g, const float* b,
                           float* __restrict__ y, int total) {
  int i = blockIdx.x * blockDim.x + threadIdx.x;
  if (i >= total) return;
  int f = i % NFEAT;
  y[i] = (x[i] - rm[f]) * rsqrtf(rv[f] + EPSN) * g[f] + b[f];
}

// LayerNorm over last dim N (block = N threads, grid = rows). Optional
// residual added pre-norm; act: 0 none, 1 gelu.
__global__ void k_ln(const float* __restrict__ X, const float* __restrict__ R,
                     const float* g, const float* b, float* __restrict__ Y,
                     int N, int act) {
  extern __shared__ float sm[];
  float* s1 = sm;
  float* s2 = sm + blockDim.x;
  const int row = blockIdx.x, tid = threadIdx.x;
  float v = X[(size_t)row * N + tid];
  if (R) v += R[(size_t)row * N + tid];
  s1[tid] = v;
  s2[tid] = v * v;
  __syncthreads();
  for (int off = blockDim.x >> 1; off > 0; off >>= 1) {
    if (tid < off) { s1[tid] += s1[tid + off]; s2[tid] += s2[tid + off]; }
    __syncthreads();
  }
  const float mean = s1[0] / (float)N;
  const float var = s2[0] / (float)N - mean * mean;
  float y = (v - mean) * rsqrtf(var + EPSN) * g[tid] + b[tid];
  if (act == 1) y = gelu_f(y);
  Y[(size_t)row * N + tid] = y;
}

// conv epilogue: X slice [*, colOff:colOff+64] -> (+bias, BN, GELU) in place
__global__ void k_bn_gelu(float* __restrict__ X, int colOff, const float* cb,
                          const float* rm, const float* rv, const float* g,
                          const float* bb, int Mrows) {
  int i = blockIdx.x * blockDim.x + threadIdx.x;
  if (i >= Mrows * 64) return;
  int row = i / 64, c = i % 64;
  float* p = &X[(size_t)row * 256 + colOff + c];
  float y = *p + cb[c];
  y = (y - rm[c]) * rsqrtf(rv[c] + EPSN) * g[c] + bb[c];
  *p = gelu_f(y);
}

// softmax over 4 logits, scale each 64-wide branch of att_in
__global__ void k_fuse4(const float* __restrict__ logits,
                        const float* __restrict__ att, float* __restrict__ out,
                        int Mrows) {
  int i = blockIdx.x * blockDim.x + threadIdx.x;
  if (i >= Mrows * 256) return;
  int row = i / 256, c = i % 256, br = c / 64;
  const float* L = &logits[(size_t)row * 4];
  float mx = fmaxf(fmaxf(L[0], L[1]), fmaxf(L[2], L[3]));
  float e0 = expf(L[0] - mx), e1 = expf(L[1] - mx), e2 = expf(L[2] - mx),
        e3 = expf(L[3] - mx);
  float den = e0 + e1 + e2 + e3;
  float w = (br == 0 ? e0 : br == 1 ? e1 : br == 2 ? e2 : e3) / den;
  out[i] = att[i] * w;
}

__global__ void k_addpos(const float* __restrict__ h, const float* __restrict__ pos,
                         float* __restrict__ y, int total) {
  int i = blockIdx.x * blockDim.x + threadIdx.x;
  if (i >= total) return;
  int c = i % HDIM;
  int s = (i / HDIM) % SEQ;
  y[i] = h[i] + pos[s * HDIM + c];
}

// fused decayed-softmax attention for one (batch, head); flash-style online
// softmax; K/V tiles staged in LDS (2*12.8KB << 320KB/WGP).
__global__ void k_attn(const float* __restrict__ q, const float* __restrict__ k,
                       const float* __restrict__ v, const float* __restrict__ dp,
                       float* __restrict__ ctx) {
  __shared__ float ks[SEQ * HEADDIM];
  __shared__ float vs[SEQ * HEADDIM];
  __shared__ float dec[SEQ];
  const int b = blockIdx.x / NHEADS;
  const int hd = blockIdx.x % NHEADS;
  const float* kb = k + (size_t)b * SEQ * HDIM + hd * HEADDIM;
  const float* vb = v + (size_t)b * SEQ * HDIM + hd * HEADDIM;
  for (int idx = threadIdx.x; idx < SEQ * HEADDIM; idx += blockDim.x) {
    int t = idx / HEADDIM, j = idx % HEADDIM;
    ks[idx] = kb[(size_t)t * HDIM + j];
    vs[idx] = vb[(size_t)t * HDIM + j];
  }
  const float dval = dp[0];
  for (int t = threadIdx.x; t < SEQ; t += blockDim.x)
    dec[t] = powf(dval, (float)(SEQ - 1 - t));
  __syncthreads();
  const int s = threadIdx.x;
  if (s < SEQ) {
    float qr[HEADDIM];
    const float* qb = q + ((size_t)b * SEQ + s) * HDIM + hd * HEADDIM;
#pragma unroll
    for (int j = 0; j < HEADDIM; ++j) qr[j] = qb[j];
    const float scale = 0.17677669529663687f;  // 1/sqrt(32)
    float m = -1e30f, l = 0.0f, acc[HEADDIM];
#pragma unroll
    for (int j = 0; j < HEADDIM; ++j) acc[j] = 0.0f;
    for (int t = 0; t < SEQ; ++t) {
      float sc = 0.0f;
#pragma unroll
      for (int j = 0; j < HEADDIM; ++j) sc += qr[j] * ks[t * HEADDIM + j];
      sc = sc * scale * dec[t];
      const float mn = fmaxf(m, sc);
      const float corr = expf(m - mn);
      const float w = expf(sc - mn);
      l = l * corr + w;
#pragma unroll
      for (int j = 0; j < HEADDIM; ++j)
        acc[j] = acc[j] * corr + w * vs[t * HEADDIM + j];
      m = mn;
    }
    const float inv = 1.0f / l;
    float* cb = ctx + ((size_t)b * SEQ + s) * HDIM + hd * HEADDIM;
#pragma unroll
    for (int j = 0; j < HEADDIM; ++j) cb[j] = acc[j] * inv;
  }
}

__global__ void k_vecadd(const float* a, const float* b, float* c, int n) {
  int i = blockIdx.x * blockDim.x + threadIdx.x;
  if (i < n) c[i] = a[i] + b[i];
}
__global__ void k_zero(float* p, int n) {
  int i = blockIdx.x * blockDim.x + threadIdx.x;
  if (i < n) p[i] = 0.0f;
}

// LSTM cell: gates = xg[:,t,:] + (h @ Whh^T); i,f,g,o split; update h,c, emit h
__global__ void k_cell(const float* __restrict__ xg, const float* __restrict__ gr,
                       float* __restrict__ hs, float* __restrict__ cs,
                       float* __restrict__ ys, int t) {
  int idx = blockIdx.x * blockDim.x + threadIdx.x;
  if (idx >= BATCH * HDIM) return;
  int b = idx / HDIM, j = idx % HDIM;
  const float* xr = xg + ((size_t)b * SEQ + t) * 4 * HDIM;
  const float* gg = gr + (size_t)b * 4 * HDIM;
  float gi = xr[j] + gg[j];
  float gf = xr[HDIM + j] + gg[HDIM + j];
  float gc = xr[2 * HDIM + j] + gg[2 * HDIM + j];
  float go = xr[3 * HDIM + j] + gg[3 * HDIM + j];
  float c = sigm_f(gf) * cs[idx] + sigm_f(gi) * tanhf(gc);
  float h = sigm_f(go) * tanhf(c);
  cs[idx] = c;
  hs[idx] = h;
  ys[((size_t)b * SEQ + t) * HDIM + j] = h;
}

// repack conv weight (O=64, I=256, k) -> (k, I=256, O=64) for GEMM-per-tap
__global__ void k_repack(const float* __restrict__ W, float* __restrict__ Wr,
                         int klen) {
  int i = blockIdx.x * blockDim.x + threadIdx.x;
  if (i >= 64 * 256 * klen) return;
  int o = i / (256 * klen);
  int rem = i % (256 * klen);
  int ic = rem / klen;
  int t = rem % klen;
  Wr[((size_t)t * 256 + ic) * 64 + o] = W[i];
}

__global__ void k_pool(const float* __restrict__ h, float* __restrict__ pooled) {
  int idx = blockIdx.x * blockDim.x + threadIdx.x;
  if (idx >= BATCH * HDIM) return;
  int b = idx / HDIM, j = idx % HDIM;
  float sum = 0.0f, mx = -1e30f, last = 0.0f;
  for (int s = 0; s < SEQ; ++s) {
    float v = h[((size_t)b * SEQ + s) * HDIM + j];
    sum += v;
    mx = fmaxf(mx, v);
    if (s == SEQ - 1) last = v;
  }
  pooled[(size_t)b * 768 + j] = sum / (float)SEQ;
  pooled[(size_t)b * 768 + 256 + j] = mx;
  pooled[(size_t)b * 768 + 512 + j] = last;
}

__global__ void k_final(const float* __restrict__ pooled, const float* __restrict__ g,
                        float* __restrict__ fin) {
  int i = blockIdx.x * blockDim.x + threadIdx.x;
  if (i >= BATCH * 768) return;
  int b = i / 768, c = i % 768;
  float v = pooled[i];
  if (c >= 512) v *= g[(size_t)b * HDIM + (c - 512)];
  fin[i] = v;
}

// ---------------------------------------------------------------------------
static inline int ceilDiv(int a, int b) { return (a + b - 1) / b; }

struct GroupInfo { int off, d, base; };

extern "C" void kernel_launch(void* const* d_in, const int* in_sizes, int n_in,
                              void* d_out, int out_size, void* d_ws, size_t ws_size,
                              hipStream_t stream) {
  (void)out_size; (void)ws_size;
  // ---- input mapping (jax pytree sorted-key flatten of params; x detected) --
  int pb;
  const float* xin;
  if (in_sizes[0] == BATCH * SEQ * NFEAT) { xin = (const float*)d_in[0]; pb = 1; }
  else { xin = (const float*)d_in[n_in - 1]; pb = 0; }
  auto P = [&](int i) { return (const float*)d_in[pb + i]; };

  // ---- workspace layout (floats) ----
  float* ws = (float*)d_ws;
  const size_t MSH = (size_t)MSROWS * HDIM;     // 6,553,600
  float* bufH = ws;
  float* bufT1 = bufH + MSH;
  float* bufT2 = bufT1 + MSH;
  float* bufBig = bufT2 + MSH;                  // 25600*1024
  float* sm = bufBig + (size_t)MSROWS * 1024;
  float* gatesR = sm;                sm += 256 * 1024;
  float* hstate = sm;                sm += BATCH * HDIM;
  float* cstate = sm;                sm += BATCH * HDIM;  // contiguous after hstate
  float* bsum = sm;                  sm += 1024;
  float* pooled = sm;                sm += BATCH * 768;
  float* gbuf = sm;                  sm += BATCH * HDIM;
  float* fin = sm;                   sm += BATCH * 768;
  float* o1 = sm;                    sm += BATCH * 256;
  float* o2 = sm;                    sm += BATCH * 128;
  float* cf1 = sm;                   sm += BATCH * 128;
  float* wr = sm;                    sm += 15 * 256 * 64;

  auto EL = [&](long long n) { return dim3((unsigned)((n + 255) / 256)); };
  auto gemm = [&](const float* A, const float* Bm, const float* bias, float* C,
                  int M, int N, int K, int lda, int ldb, int ldc, int opB,
                  int accum, int act, int shiftS = 0, int seqLen = 0) {
    int tiles = ceilDiv(M, 16) * ceilDiv(N, 16);
    k_gemm<<<dim3(ceilDiv(tiles, 4)), dim3(128), 0, stream>>>(
        A, Bm, bias, C, M, N, K, lda, ldb, ldc, opB, accum, act, shiftS, seqLen);
  };
  auto ln = [&](const float* X, const float* R, const float* g_, const float* b_,
                float* Y, int Mrows, int N, int act) {
    k_ln<<<dim3(Mrows), dim3(N), 2 * N * sizeof(float), stream>>>(X, R, g_, b_, Y, N, act);
  };

  // ---- param index bases (sorted-key flatten) ----
  // attn[a]: a*10 + {Wk0,Wo1,Wq2,Wv3,bk4,bo5,bq6,bv7,decay8,pos9}
  // conf: 20..23 {W1,W2,b1,b2}; conv1: 24 (+30 -> conv2: 54)
  // fn: 84..87 {b,g,rm,rv}; gate: 88..93 {W1,W2,b1,b2,lnb,lng}
  // lstm[i]: 94+i*10 {Whh,Wih,bhh,bih}x2, lnb+8, lng+9
  // out: 124..133 {W1,W2,W3,b1,b2,b3,lnb1,lnb2,lng1,lng2}
  // proj sorted names 134..181; projfinal: 182..185 {W,b,lnb,lng}
  static const GroupInfo groups[8] = {
      {0, 5, 152},  {5, 7, 176},  {12, 11, 170}, {23, 14, 146},
      {37, 5, 140}, {42, 10, 158}, {52, 10, 164}, {62, 12, 134}};

  // 1) feature norm -> bufT1 (MS x 74)
  k_featnorm<<<EL((long long)MSROWS * NFEAT), 256, 0, stream>>>(
      xin, P(86), P(87), P(85), P(84), bufT1, MSROWS * NFEAT);

  // 2) per-group projections -> feats in bufBig (MS x 512)
  for (int gi = 0; gi < 8; ++gi) {
    const GroupInfo& G = groups[gi];
    gemm(bufT1 + G.off, P(G.base + 0), P(G.base + 2), bufT2, MSROWS, 64, G.d,
         NFEAT, 64, 64, 0, 0, 0);
    ln(bufT2, nullptr, P(G.base + 5), P(G.base + 4), bufT2, MSROWS, 64, 1);
    gemm(bufT2, P(G.base + 1), P(G.base + 3), bufBig + gi * 64, MSROWS, 64, 64,
         64, 64, 512, 0, 0, 0);
  }

  // 3) projfinal 512->256, LN+GELU -> bufH
  gemm(bufBig, P(182), P(183), bufT1, MSROWS, 256, 512, 512, 256, 256, 0, 0, 0);
  ln(bufT1, nullptr, P(185), P(184), bufH, MSROWS, 256, 1);

  // 4) conv blocks (taps as shifted GEMMs)
  const int KS1[4] = {3, 5, 7, 11};
  const int KS2[4] = {5, 7, 11, 15};
  for (int cbi = 0; cbi < 2; ++cbi) {
    const int cb = (cbi == 0) ? 24 : 54;
    const int* ks = (cbi == 0) ? KS1 : KS2;
    for (int c = 0; c < 4; ++c) {
      const int k = ks[c];
      const int dil = (k <= 5) ? 1 : 2;
      const int pad = (k - 1) * dil / 2;
      k_repack<<<EL((long long)64 * 256 * k), 256, 0, stream>>>(P(cb + c * 6 + 0), wr, k);
      for (int t = 0; t < k; ++t)
        gemm(bufH, wr + (size_t)t * 256 * 64, nullptr, bufBig + c * 64, MSROWS,
             64, 256, 256, 64, 256, 0, (t > 0) ? 1 : 0, 0, t * dil - pad, SEQ);
      k_bn_gelu<<<EL((long long)MSROWS * 64), 256, 0, stream>>>(
          bufBig, c * 64, P(cb + c * 6 + 1), P(cb + c * 6 + 4), P(cb + c * 6 + 5),
          P(cb + c * 6 + 3), P(cb + c * 6 + 2), MSROWS);
    }
    gemm(bufBig, P(cb + 24), P(cb + 26), bufT2, MSROWS, 64, 256, 256, 64, 64, 0, 0, 1);
    float* logits = bufT2 + (size_t)MSROWS * 64;
    gemm(bufT2, P(cb + 25), P(cb + 27), logits, MSROWS, 4, 64, 64, 4, 4, 0, 0, 0);
    k_fuse4<<<EL((long long)MSROWS * 256), 256, 0, stream>>>(logits, bufBig, bufT1, MSROWS);
    ln(bufT1, nullptr, P(cb + 29), P(cb + 28), bufH, MSROWS, 256, 0);
  }

  // 5) LSTM blocks (3 x 2 layers, 100 sequential steps each)
  for (int blk = 0; blk < 3; ++blk) {
    const int lbase = 94 + blk * 10;
    const float* cur = bufH;
    float* outbuf[2] = {bufT1, bufT2};
    for (int l = 0; l < 2; ++l) {
      const int lb2 = lbase + l * 4;  // Whh+0, Wih+1, bhh+2, bih+3
      k_vecadd<<<EL(1024), 256, 0, stream>>>(P(lb2 + 3), P(lb2 + 2), bsum, 1024);
      gemm(cur, P(lb2 + 1), bsum, bufBig, MSROWS, 1024, 256, 256, 256, 1024, 1, 0, 0);
      k_zero<<<EL(2 * BATCH * HDIM), 256, 0, stream>>>(hstate, 2 * BATCH * HDIM);
      for (int t = 0; t < SEQ; ++t) {
        gemm(hstate, P(lb2 + 0), nullptr, gatesR, BATCH, 1024, 256, 256, 256, 1024, 1, 0, 0);
        k_cell<<<EL(BATCH * HDIM), 256, 0, stream>>>(bufBig, gatesR, hstate,
                                                     cstate, outbuf[l], t);
      }
      cur = outbuf[l];
    }
    ln(bufT2, bufH, P(lbase + 9), P(lbase + 8), bufH, MSROWS, 256, 0);
  }

  // 6) attention blocks (residual)
  for (int a = 0; a < 2; ++a) {
    const int ab = a * 10;
    k_addpos<<<EL((long long)MSROWS * HDIM), 256, 0, stream>>>(bufH, P(ab + 9),
                                                               bufT1, MSROWS * HDIM);
    float* q = bufBig;
    float* kk = bufBig + MSH;
    float* v = bufBig + 2 * MSH;
    gemm(bufT1, P(ab + 2), P(ab + 6), q, MSROWS, 256, 256, 256, 256, 256, 0, 0, 0);
    gemm(bufT1, P(ab + 0), P(ab + 4), kk, MSROWS, 256, 256, 256, 256, 256, 0, 0, 0);
    gemm(bufH, P(ab + 3), P(ab + 7), v, MSROWS, 256, 256, 256, 256, 256, 0, 0, 0);
    k_attn<<<dim3(BATCH * NHEADS), dim3(128), 0, stream>>>(q, kk, v, P(ab + 8), bufT1);
    gemm(bufT1, P(ab + 1), P(ab + 5), bufH, MSROWS, 256, 256, 256, 256, 256, 0, 1, 0);
  }

  // 7) pooling, gate, final
  k_pool<<<EL(BATCH * HDIM), 256, 0, stream>>>(bufH, pooled);
  gemm(pooled, P(88), P(90), o1, BATCH, 256, 768, 768, 256, 256, 0, 0, 0);
  ln(o1, nullptr, P(93), P(92), o1, BATCH, 256, 1);
  gemm(o1, P(89), P(91), gbuf, BATCH, 256, 256, 256, 256, 256, 0, 0, 2);
  k_final<<<EL(BATCH * 768), 256, 0, stream>>>(pooled, gbuf, fin);

  // 8) output head -> d_out[0:768]
  float* dout = (float*)d_out;
  gemm(fin, P(124), P(127), o1, BATCH, 256, 768, 768, 256, 256, 0, 0, 0);
  ln(o1, nullptr, P(132), P(130), o1, BATCH, 256, 1);
  gemm(o1, P(125), P(128), o2, BATCH, 128, 256, 256, 128, 128, 0, 0, 0);
  ln(o2, nullptr, P(133), P(131), o2, BATCH, 128, 1);
  gemm(o2, P(126), P(129), dout, BATCH, 3, 128, 128, 3, 3, 0, 0, 0);

  // 9) confidence head -> d_out[768:1024]
  gemm(fin, P(20), P(22), cf1, BATCH, 128, 768, 768, 128, 128, 0, 0, 1);
  gemm(cf1, P(21), P(23), dout + BATCH * 3, BATCH, 1, 128, 128, 1, 1, 0, 0, 2);
}